// find_median_52286931861979
// MI455X (gfx1250) — compile-verified
//
#include <hip/hip_runtime.h>

// Reference: x[T=64][C=16][D=65536] = input*k ; 7x soft-removal along C;
// m1=logsumexp; 1x soft-removal; m2=logsumexp; out = (m1+m2)/2/k -> [T, D] fp32.
// No GEMM anywhere => WMMA inapplicable; trans/VALU-bound columnwise kernel.
//
// Base-2 trick: with y = x*log2(e), softmax_e(x) == softmax_2(y) and the
// soft-removal update is linear, so the whole iteration runs on y using raw
// v_exp_f32 (natively exp2) with no per-exp multiplies. logsumexp_e recovers
// as ln2*(max(y) + log2(sum)).

typedef float v4f __attribute__((ext_vector_type(4)));

#define T_SPAN 64
#define CROWS  16
#define KSCALE 100.0f
#define LOG2E  1.44269504088896340736f
#define LN2    0.69314718055994530942f

#if __has_builtin(__builtin_amdgcn_exp2f)
#define FAST_EXP2(v) __builtin_amdgcn_exp2f(v)
#else
#define FAST_EXP2(v) exp2f(v)
#endif
#if __has_builtin(__builtin_amdgcn_logf)
#define FAST_LOG2(v) __builtin_amdgcn_logf(v)
#else
#define FAST_LOG2(v) log2f(v)
#endif
#if __has_builtin(__builtin_amdgcn_rcpf)
#define FAST_RCP(v) __builtin_amdgcn_rcpf(v)
#else
#define FAST_RCP(v) (1.0f / (v))
#endif

static __device__ __forceinline__ v4f v4_max(v4f a, v4f b) {
  v4f r;
  r.x = fmaxf(a.x, b.x);
  r.y = fmaxf(a.y, b.y);
  r.z = fmaxf(a.z, b.z);
  r.w = fmaxf(a.w, b.w);
  return r;
}

static __device__ __forceinline__ v4f v4_exp2(v4f a) {
  v4f r;
  r.x = FAST_EXP2(a.x);
  r.y = FAST_EXP2(a.y);
  r.z = FAST_EXP2(a.z);
  r.w = FAST_EXP2(a.w);
  return r;
}

static __device__ __forceinline__ v4f v4_log2(v4f a) {
  v4f r;
  r.x = FAST_LOG2(a.x);
  r.y = FAST_LOG2(a.y);
  r.z = FAST_LOG2(a.z);
  r.w = FAST_LOG2(a.w);
  return r;
}

static __device__ __forceinline__ v4f v4_rcp(v4f a) {
  v4f r;
  r.x = FAST_RCP(a.x);
  r.y = FAST_RCP(a.y);
  r.z = FAST_RCP(a.z);
  r.w = FAST_RCP(a.w);
  return r;
}

// Balanced max tree over the 16 rows (short dep chain; backend can form max3).
static __device__ __forceinline__ v4f v4_max16(const v4f* y) {
  v4f m01 = v4_max(y[0], y[1]),   m23 = v4_max(y[2], y[3]);
  v4f m45 = v4_max(y[4], y[5]),   m67 = v4_max(y[6], y[7]);
  v4f m89 = v4_max(y[8], y[9]),   mab = v4_max(y[10], y[11]);
  v4f mcd = v4_max(y[12], y[13]), mef = v4_max(y[14], y[15]);
  v4f a = v4_max(v4_max(m01, m23), v4_max(m45, m67));
  v4f b = v4_max(v4_max(m89, mab), v4_max(mcd, mef));
  return v4_max(a, b);
}

__global__ __launch_bounds__(256) void median_softremove_kernel(
    const float* __restrict__ in, float* __restrict__ out, int d4count) {
  const int t  = blockIdx.y;                      // time chunk 0..63
  const int d4 = blockIdx.x * 256 + threadIdx.x;  // group of 4 columns
  if (d4 >= d4count) return;

  const v4f* __restrict__ src =
      (const v4f*)in + (size_t)t * CROWS * d4count + d4;

  // Load the 16-row column group (512B contiguous per wave per load),
  // non-temporal (256MB streamed exactly once), pre-scaled into base-2 domain.
  v4f y[CROWS];
#pragma unroll
  for (int c = 0; c < CROWS; ++c) {
    v4f v = __builtin_nontemporal_load(src + (size_t)c * d4count);
    y[c] = v * (KSCALE * LOG2E);
  }

  v4f e[CROWS];

  // 7 soft-removal iterations: y <- y - softmax2(y)*y (stable, base-2).
#pragma unroll 1
  for (int it = 0; it < 7; ++it) {
    v4f m = v4_max16(y);
    v4f s = {0.0f, 0.0f, 0.0f, 0.0f};
#pragma unroll
    for (int c = 0; c < CROWS; ++c) {
      e[c] = v4_exp2(y[c] - m);
      s += e[c];
    }
    v4f rs = v4_rcp(s);
#pragma unroll
    for (int c = 0; c < CROWS; ++c) {
      y[c] = y[c] - y[c] * (e[c] * rs);  // y *= (1 - softmax)
    }
  }

  // m1 = logsumexp and the 8th removal share the same exp2(y-m) and sum.
  v4f m1m = v4_max16(y);
  v4f s1 = {0.0f, 0.0f, 0.0f, 0.0f};
#pragma unroll
  for (int c = 0; c < CROWS; ++c) {
    e[c] = v4_exp2(y[c] - m1m);
    s1 += e[c];
  }
  v4f lse1 = m1m + v4_log2(s1);  // logsumexp in base-2 domain
  v4f rs = v4_rcp(s1);
#pragma unroll
  for (int c = 0; c < CROWS; ++c) {
    y[c] = y[c] - y[c] * (e[c] * rs);
  }

  // m2 = logsumexp of the updated column.
  v4f m2m = v4_max16(y);
  v4f s2 = {0.0f, 0.0f, 0.0f, 0.0f};
#pragma unroll
  for (int c = 0; c < CROWS; ++c) {
    s2 += v4_exp2(y[c] - m2m);
  }
  v4f lse2 = m2m + v4_log2(s2);

  // out = (ln2*lse1 + ln2*lse2) / (2*K)
  v4f res = (lse1 + lse2) * (LN2 * 0.5f / KSCALE);
  __builtin_nontemporal_store(res, (v4f*)out + (size_t)t * d4count + d4);
}

extern "C" void kernel_launch(void* const* d_in, const int* in_sizes, int n_in,
                              void* d_out, int out_size, void* d_ws, size_t ws_size,
                              hipStream_t stream) {
  (void)d_ws; (void)ws_size; (void)n_in; (void)in_sizes;
  const float* in = (const float*)d_in[0];
  float* out = (float*)d_out;

  const int D = out_size / T_SPAN;  // 65536 for the reference shapes
  const int d4count = D / 4;        // 16384 (D is a multiple of 4)

  dim3 block(256);
  dim3 grid((d4count + 255) / 256, T_SPAN);
  median_softremove_kernel<<<grid, block, 0, stream>>>(in, out, d4count);
}